// TextureNetworkBaseline_14396730376509
// MI455X (gfx1250) — compile-verified
//
#include <hip/hip_runtime.h>
#include <math.h>

#define B_    4
#define NR_   2
#define NREF_ 4
#define H_    320
#define W_    320
#define HW_   (H_ * W_)

typedef __attribute__((ext_vector_type(2))) float v2f;
typedef __attribute__((ext_vector_type(8))) float v8f;
typedef __attribute__((ext_vector_type(4))) unsigned int u32x4;
typedef __attribute__((ext_vector_type(4))) int i32x4;
typedef __attribute__((ext_vector_type(8))) int i32x8;

#if defined(__HIP_DEVICE_COMPILE__) && __has_builtin(__builtin_amdgcn_tensor_load_to_lds)
#define USE_TDM 1
#else
#define USE_TDM 0
#endif

// ---------- small 3x3 helpers (setup kernel only) ----------
__device__ inline void inv3x3(const float m[9], float o[9]) {
    float a = m[0], b = m[1], c = m[2];
    float d = m[3], e = m[4], f = m[5];
    float g = m[6], h = m[7], i = m[8];
    float A =  (e * i - f * h);
    float Bc = -(d * i - f * g);
    float C =  (d * h - e * g);
    float det = a * A + b * Bc + c * C;
    float inv = 1.0f / det;
    o[0] = A * inv;   o[1] = -(b * i - c * h) * inv;  o[2] =  (b * f - c * e) * inv;
    o[3] = Bc * inv;  o[4] =  (a * i - c * g) * inv;  o[5] = -(a * f - c * d) * inv;
    o[6] = C * inv;   o[7] = -(a * h - b * g) * inv;  o[8] =  (a * e - b * d) * inv;
}
__device__ inline void mm33(const float* A, const float* Bm, float* C) {
    for (int r = 0; r < 3; r++)
        for (int c = 0; c < 3; c++)
            C[3 * r + c] = A[3 * r] * Bm[c] + A[3 * r + 1] * Bm[3 + c] + A[3 * r + 2] * Bm[6 + c];
}
__device__ inline void mv33(const float* A, const float* v, float* o) {
    for (int r = 0; r < 3; r++)
        o[r] = A[3 * r] * v[0] + A[3 * r + 1] * v[1] + A[3 * r + 2] * v[2];
}

// Precompute per-(render cam, ref cam) combined projection matrices P (3x4):
//   pix_homog = P * (x*d, y*d, d, 1)
// Stored padded to 16x4 per (b,r), in the exact WMMA A-operand row layout:
//   row m = 4*ref + comp (comp 3 = zero pad row).
__global__ void setup_proj(const float* __restrict__ camK,  const float* __restrict__ camW,
                           const float* __restrict__ camKr, const float* __restrict__ camWr,
                           float* __restrict__ P) {
    int t = threadIdx.x;
    if (t >= B_ * NR_ * NREF_) return;
    int ref = t % NREF_;
    int br  = t / NREF_;
    int b   = br / NR_;

    const float* K  = camK  + br * 9;
    const float* Wm = camW  + br * 12;
    const float* Kr = camKr + (b * NREF_ + ref) * 9;
    const float* Wr = camWr + (b * NREF_ + ref) * 12;

    float R[9], tv[3];
    for (int r = 0; r < 3; r++) {
        for (int c = 0; c < 3; c++) R[3 * r + c] = Wm[4 * r + c];
        tv[r] = Wm[4 * r + 3];
    }
    float invK[9], invR[9];
    inv3x3(K, invK);
    inv3x3(R, invR);

    // p3d = -invR*invK*(xd,yd,d) - invR*t   (cube_diagonal divide*multiply cancels)
    float M3[9];  mm33(invR, invK, M3);
    for (int i = 0; i < 9; i++) M3[i] = -M3[i];
    float c3[3];  mv33(invR, tv, c3);
    for (int i = 0; i < 3; i++) c3[i] = -c3[i];

    float Rr[9], tr[3];
    for (int r = 0; r < 3; r++) {
        for (int c = 0; c < 3; c++) Rr[3 * r + c] = Wr[4 * r + c];
        tr[r] = Wr[4 * r + 3];
    }
    float KR[9];  mm33(Kr, Rr, KR);
    float PL[9];  mm33(KR, M3, PL);
    float Rc[3];  mv33(Rr, c3, Rc);
    for (int i = 0; i < 3; i++) Rc[i] += tr[i];
    float PT[3];  mv33(Kr, Rc, PT);

    float* dst = P + br * 64;   // 16 rows x 4 cols
    for (int comp = 0; comp < 3; comp++) {
        int m = 4 * ref + comp;
        dst[m * 4 + 0] = PL[3 * comp + 0];
        dst[m * 4 + 1] = PL[3 * comp + 1];
        dst[m * 4 + 2] = PL[3 * comp + 2];
        dst[m * 4 + 3] = PT[comp];
    }
    int mz = 4 * ref + 3;
    dst[mz * 4 + 0] = 0.0f; dst[mz * 4 + 1] = 0.0f;
    dst[mz * 4 + 2] = 0.0f; dst[mz * 4 + 3] = 0.0f;
}

// One wave32 handles 16 pixels:
//   A (16x4) = 4 stacked 3x4 ref projections (TDM-staged in LDS)
//   B (4x16) = 16 homogeneous pixel vectors (xd, yd, d, 1)
//   D = WMMA(A,B): lanes 0-15 get refs {0,1}, lanes 16-31 get refs {2,3}
__global__ void __launch_bounds__(256) texwarp(
        const float* __restrict__ depth,
        const float* __restrict__ image_ref,
        const float* __restrict__ background,
        const float* __restrict__ Pg,
        float* __restrict__ out) {
    const int lane = threadIdx.x & 31;
    const int gw   = blockIdx.x * 8 + (threadIdx.x >> 5);
    const int tilesPerImg = HW_ / 16;              // 6400; divisible by 8 -> block-uniform br
    const int br  = gw / tilesPerImg;
    const int b   = br / NR_;
    const int pix = (gw % tilesPerImg) * 16 + (lane & 15);
    const int y   = pix / W_;
    const int x   = pix % W_;

#if USE_TDM
    // Stage the 16x4 A-matrix tile (256 B) into LDS with the Tensor Data Mover.
    __shared__ float Plds[64];
    if (threadIdx.x == 0) {
        const int brB = (blockIdx.x * 8) / tilesPerImg;
        unsigned long long ga = (unsigned long long)(uintptr_t)(Pg + brB * 64);
        unsigned int ldsoff   = (unsigned int)(uintptr_t)(&Plds[0]);
        u32x4 g0;
        g0[0] = 1u;                                              // count=1 (valid user D#)
        g0[1] = ldsoff;                                          // lds_addr
        g0[2] = (unsigned int)ga;                                // global_addr[31:0]
        g0[3] = (unsigned int)((ga >> 32) & 0x01FFFFFFu)         // global_addr[56:32]
              | (2u << 30);                                      // type = 2 ("image")
        i32x8 g1;
        g1[0] = (int)(2u << 16);     // workgroup_mask=0, data_size=2 (4 bytes)
        g1[1] = (int)(64u << 16);    // tensor_dim0 = 64 elements (bits 79:48, low half)
        g1[2] = (int)(1u << 16);     // tensor_dim1 = 1 (bits 111:80, low half)
        g1[3] = (int)(64u << 16);    // tile_dim0 = 64 (bits 127:112)
        g1[4] = 1;                   // tile_dim1 = 1
        g1[5] = 64;                  // tensor_dim0_stride = 64
        g1[6] = 0;
        g1[7] = 0;
        i32x4 gz4 = {0, 0, 0, 0};
#if defined(__clang_major__) && (__clang_major__ >= 23)
        i32x8 gz8 = {0, 0, 0, 0, 0, 0, 0, 0};
        __builtin_amdgcn_tensor_load_to_lds(g0, g1, gz4, gz4, gz8, 0);
#else
        __builtin_amdgcn_tensor_load_to_lds(g0, g1, gz4, gz4, 0);
#endif
        __builtin_amdgcn_s_wait_tensorcnt(0);
    }
    __syncthreads();
#endif

    // A operand: lane l holds row m=l&15, K-pair (l>>4)*2
    const int m  = lane & 15;
    const int kb = (lane >> 4) * 2;
    v2f a;
#if USE_TDM
    a.x = Plds[m * 4 + kb];
    a.y = Plds[m * 4 + kb + 1];
#else
    a.x = Pg[br * 64 + m * 4 + kb];
    a.y = Pg[br * 64 + m * 4 + kb + 1];
#endif

    // B operand: VGPR r holds rows {r, r+2}. Branchless (EXEC all-ones into WMMA).
    const float d  = depth[(size_t)br * HW_ + pix];
    const float bx = (float)x * d;
    const float by = (float)y * d;
    v2f bv;
    bv.x = (lane < 16) ? bx : d;
    bv.y = (lane < 16) ? by : 1.0f;

    v8f acc = {};
    acc = __builtin_amdgcn_wmma_f32_16x16x4_f32(false, a, false, bv,
                                                (short)0, acc, false, false);
    // lane<16 : acc[0..2]=ref0 (hx,hy,hz), acc[4..6]=ref1
    // lane>=16: acc[0..2]=ref2,            acc[4..6]=ref3

    float sumR = 0.0f, sumG = 0.0f, sumB = 0.0f;
    if (d > 0.0f) {
        const int refBase = (lane < 16) ? 0 : 2;
#pragma unroll
        for (int j = 0; j < 2; j++) {
            const float hz = acc[4 * j + 2];
            const float rz = __builtin_amdgcn_rcpf(hz);      // ~1 ulp: error << 1e-4 px
            const float px = acc[4 * j + 0] * rz;
            const float py = acc[4 * j + 1] * rz;
            const float* img = image_ref + (size_t)(b * NREF_ + refBase + j) * 3 * HW_;

            const float x0f = floorf(px), y0f = floorf(py);
            const float wx = px - x0f,    wy = py - y0f;
            const float x1f = x0f + 1.0f, y1f = y0f + 1.0f;

            const bool vx0 = (x0f >= 0.0f) && (x0f <= (float)(W_ - 1));
            const bool vx1 = (x1f >= 0.0f) && (x1f <= (float)(W_ - 1));
            const bool vy0 = (y0f >= 0.0f) && (y0f <= (float)(H_ - 1));
            const bool vy1 = (y1f >= 0.0f) && (y1f <= (float)(H_ - 1));

            // clamped indices -> loads always legal; validity folded into weights
            const int ix0 = (int)fminf(fmaxf(x0f, 0.0f), (float)(W_ - 1));
            const int ix1 = (int)fminf(fmaxf(x1f, 0.0f), (float)(W_ - 1));
            const int iy0 = (int)fminf(fmaxf(y0f, 0.0f), (float)(H_ - 1));
            const int iy1 = (int)fminf(fmaxf(y1f, 0.0f), (float)(H_ - 1));

            const float w00 = (vx0 && vy0) ? (1.0f - wx) * (1.0f - wy) : 0.0f;
            const float w01 = (vx1 && vy0) ? wx * (1.0f - wy)          : 0.0f;
            const float w10 = (vx0 && vy1) ? (1.0f - wx) * wy          : 0.0f;
            const float w11 = (vx1 && vy1) ? wx * wy                   : 0.0f;

            const int i00 = iy0 * W_ + ix0, i01 = iy0 * W_ + ix1;
            const int i10 = iy1 * W_ + ix0, i11 = iy1 * W_ + ix1;

            sumR += w00 * img[i00]            + w01 * img[i01]
                  + w10 * img[i10]            + w11 * img[i11];
            sumG += w00 * img[i00 + HW_]      + w01 * img[i01 + HW_]
                  + w10 * img[i10 + HW_]      + w11 * img[i11 + HW_];
            sumB += w00 * img[i00 + 2 * HW_]  + w01 * img[i01 + 2 * HW_]
                  + w10 * img[i10 + 2 * HW_]  + w11 * img[i11 + 2 * HW_];
        }
    }
    // fold refs {2,3} (upper half lanes) into refs {0,1} (lower half lanes)
    sumR += __shfl_xor(sumR, 16, 32);
    sumG += __shfl_xor(sumG, 16, 32);
    sumB += __shfl_xor(sumB, 16, 32);

    if (lane < 16) {
        const size_t o = (size_t)br * 3 * HW_ + (size_t)pix;
        if (d > 0.0f) {
            out[o]           = sumR * 0.25f;
            out[o + HW_]     = sumG * 0.25f;
            out[o + 2 * HW_] = sumB * 0.25f;
        } else {
            out[o]           = background[o];
            out[o + HW_]     = background[o + HW_];
            out[o + 2 * HW_] = background[o + 2 * HW_];
        }
    }
}

extern "C" void kernel_launch(void* const* d_in, const int* in_sizes, int n_in,
                              void* d_out, int out_size, void* d_ws, size_t ws_size,
                              hipStream_t stream) {
    const float* depth      = (const float*)d_in[0];
    const float* camK       = (const float*)d_in[1];
    const float* camW       = (const float*)d_in[2];
    const float* image_ref  = (const float*)d_in[3];
    const float* background = (const float*)d_in[4];
    // d_in[5] = cube_diagonal: divide-then-multiply cancels exactly -> unused
    const float* camKr      = (const float*)d_in[6];
    const float* camWr      = (const float*)d_in[7];

    float* P = (float*)d_ws;   // 8 * 64 floats = 2 KB of combined projections

    setup_proj<<<1, 32, 0, stream>>>(camK, camW, camKr, camWr, P);

    const int waves  = (B_ * NR_ * HW_) / 16;   // 51200 waves, 16 px each
    const int blocks = waves / 8;               // 256 threads = 8 wave32 per block
    texwarp<<<blocks, 256, 0, stream>>>(depth, image_ref, background, P, (float*)d_out);
}